// PANPooling_68539088110413
// MI455X (gfx1250) — compile-verified
//
#include <hip/hip_runtime.h>
#include <math.h>

typedef __attribute__((ext_vector_type(2))) float v2f;
typedef __attribute__((ext_vector_type(8))) float v8f;

#define THREADS 256
#define MAX_C 512   // LDS staging capacity for p (C=128 in this problem)

#if defined(__gfx1250__) && __has_builtin(__builtin_amdgcn_wmma_f32_16x16x4_f32)
#define USE_WMMA_F32 1
#endif

__global__ void zero_f32_kernel(float* __restrict__ p, int n) {
  int i = blockIdx.x * blockDim.x + threadIdx.x;
  if (i < n) p[i] = 0.f;
}

// score2[c] += ew[e] for all edges with col[e]==c
__global__ void seg_sum_kernel(const float* __restrict__ ew, const int* __restrict__ col,
                               float* __restrict__ score2, int E) {
  int e = blockIdx.x * blockDim.x + threadIdx.x;
  if (e < E) atomicAdd(&score2[col[e]], ew[e]);
}

// s[i] = tanh((beta0*(x@p)[i] + beta1*score2[i]) * sign(w)), GEMV via f32 WMMA.
// One wave32 handles 16 rows: A tile 16x4 f32, B = p chunk replicated across
// all 16 columns so every lane's D column holds the dot product.
__global__ __launch_bounds__(THREADS) void score_wmma_kernel(
    const float* __restrict__ x, const float* __restrict__ p,
    const float* __restrict__ score2, const float* __restrict__ beta,
    const float* __restrict__ select_w, float* __restrict__ s_out,
    int nrows, int C) {
  __shared__ float p_lds[MAX_C];
  for (int i = threadIdx.x; i < C; i += blockDim.x) p_lds[i] = p[i];
  __syncthreads();                                // before any wave exits

  int lane = threadIdx.x & 31;
  int wave = blockIdx.x * (blockDim.x >> 5) + (threadIdx.x >> 5);
  int row0 = wave * 16;
  if (row0 >= nrows) return;                      // wave-uniform exit (EXEC stays full)

  float b0 = beta[0], b1 = beta[1];
  float sgn = (select_w[0] < 0.f) ? -1.f : 1.f;   // w / ||w|| for a length-1 w

#ifdef USE_WMMA_F32
  int m     = lane & 15;
  int khalf = (lane >> 4) * 2;                    // lanes 0-15: K=0,1 ; lanes 16-31: K=2,3
  int row   = row0 + m;
  if (row >= nrows) row = nrows - 1;              // clamp: no divergent guard in hot loop
  const float* xr = x + (size_t)row * C + khalf;  // 8B-aligned (C%4==0, khalf even)
  const float* pl = p_lds + khalf;

  v8f acc = {0.f, 0.f, 0.f, 0.f, 0.f, 0.f, 0.f, 0.f};
#pragma unroll 4
  for (int kc = 0; kc < C; kc += 4) {
    v2f a = *(const v2f*)(xr + kc);               // global_load_b64
    v2f b = *(const v2f*)(pl + kc);               // ds_load_b64 (replicated over N cols)
    // v_wmma_f32_16x16x4_f32: D = A(16x4) x B(4x16) + C
    acc = __builtin_amdgcn_wmma_f32_16x16x4_f32(false, a, false, b,
                                                (short)0, acc, false, false);
  }
  // C/D layout: VGPR r, lanes 0-15 -> M=r (N=lane), lanes 16-31 -> M=r+8.
  // Every column is identical, so lane 0 emits rows 0..7, lane 16 rows 8..15.
  if ((lane & 15) == 0) {
    int mbase = row0 + (lane >> 4) * 8;
#pragma unroll
    for (int r = 0; r < 8; ++r) {
      int rr = mbase + r;
      if (rr < nrows) {
        float sc = b0 * acc[r] + b1 * score2[rr];
        s_out[rr] = tanhf(sc * sgn);
      }
    }
  }
#else
  // Scalar fallback (builtin unavailable): lanes 0-15 each own one row.
  int m = lane & 15;
  int row = row0 + m;
  if (lane < 16 && row < nrows) {
    const float* xr = x + (size_t)row * C;
    float acc = 0.f;
    for (int c = 0; c < C; ++c) acc += xr[c] * p_lds[c];
    float sc = b0 * acc + b1 * score2[row];
    s_out[row] = tanhf(sc * sgn);
  }
#endif
}

// Pad sort arrays: real entries (value, index), pads (-inf, INT_MAX).
__global__ void sort_init_kernel(const float* __restrict__ s, float* __restrict__ sv,
                                 int* __restrict__ si, int N, int M) {
  int i = blockIdx.x * blockDim.x + threadIdx.x;
  if (i >= M) return;
  if (i < N) { sv[i] = s[i];         si[i] = i; }
  else       { sv[i] = -INFINITY;    si[i] = 0x7fffffff; }
}

// Global bitonic sort step, final order: value descending, index ascending on ties
// (matches jax.lax.top_k's stable tie-break).
__global__ void bitonic_step_kernel(float* __restrict__ v, int* __restrict__ idx,
                                    int j, int k, int M) {
  int i = blockIdx.x * blockDim.x + threadIdx.x;
  if (i >= M) return;
  int ixj = i ^ j;
  if (ixj > i) {
    float va = v[i], vb = v[ixj];
    int ia = idx[i], ib = idx[ixj];
    bool a_first = (va > vb) || (va == vb && ia < ib);  // "a belongs before b" (desc)
    bool desc = ((i & k) == 0);
    bool do_swap = desc ? !a_first : a_first;
    if (do_swap) {
      v[i] = vb; v[ixj] = va;
      idx[i] = ib; idx[ixj] = ia;
    }
  }
}

__global__ void node_new_init_kernel(int* __restrict__ nn, int N) {
  int i = blockIdx.x * blockDim.x + threadIdx.x;
  if (i < N) nn[i] = -1;
}

__global__ void node_new_scatter_kernel(const int* __restrict__ perm,
                                        int* __restrict__ nn, int K) {
  int i = blockIdx.x * blockDim.x + threadIdx.x;
  if (i < K) nn[perm[i]] = i;
}

// x_pool[k,c] = x[perm[k], c] * topv[k]
__global__ void gather_pool_kernel(const float* __restrict__ x,
                                   const float* __restrict__ sv, const int* __restrict__ si,
                                   float* __restrict__ out, int K, int C) {
  int i = blockIdx.x * blockDim.x + threadIdx.x;
  if (i >= K * C) return;
  int k = i / C;
  int c = i - k * C;
  out[i] = x[(size_t)si[k] * C + c] * sv[k];
}

// edge_index = [[nn[col]],[nn[row]]] if both kept else -1; ew_out masked.
__global__ void edges_kernel(const int* __restrict__ row, const int* __restrict__ col,
                             const float* __restrict__ ew, const int* __restrict__ nn,
                             float* __restrict__ out_ei, float* __restrict__ out_ew, int E) {
  int e = blockIdx.x * blockDim.x + threadIdx.x;
  if (e >= E) return;
  int pf = e + 8192; pf = pf < E ? pf : E - 1;
  __builtin_prefetch(&col[pf], 0, 1);             // global_prefetch_b8
  __builtin_prefetch(&row[pf], 0, 1);
  int a = nn[col[e]];
  int b = nn[row[e]];
  bool keep = (a >= 0) && (b >= 0);
  out_ei[e]             = (float)(keep ? a : -1);
  out_ei[(size_t)E + e] = (float)(keep ? b : -1);
  out_ew[e]             = keep ? ew[e] : 0.f;
}

__global__ void tail_kernel(const int* __restrict__ batch,
                            const float* __restrict__ sv, const int* __restrict__ si,
                            float* __restrict__ out_batch, float* __restrict__ out_perm,
                            float* __restrict__ out_topv, int K) {
  int i = blockIdx.x * blockDim.x + threadIdx.x;
  if (i >= K) return;
  int p = si[i];
  out_batch[i] = (float)batch[p];
  out_perm[i]  = (float)p;
  out_topv[i]  = sv[i];
}

static inline int grids(long long n, int t) { return (int)((n + t - 1) / t); }

extern "C" void kernel_launch(void* const* d_in, const int* in_sizes, int n_in,
                              void* d_out, int out_size, void* d_ws, size_t ws_size,
                              hipStream_t stream) {
  const float* x        = (const float*)d_in[0];
  const int*   row      = (const int*)  d_in[1];
  const int*   col      = (const int*)  d_in[2];
  const float* ew       = (const float*)d_in[3];
  const int*   batch    = (const int*)  d_in[4];
  const float* p        = (const float*)d_in[5];
  const float* beta     = (const float*)d_in[6];
  const float* select_w = (const float*)d_in[7];

  const int E = in_sizes[1];
  const int N = in_sizes[4];
  const int C = in_sizes[5];
  const int K = (N + 1) / 2;            // ceil(0.5 * N)
  int M = 1; while (M < N) M <<= 1;     // bitonic pad (131072 for N=100000)

  // workspace: s[N] | score2[N] | sortv[M] | sorti[M] | node_new[N]  (~2.25 MB)
  float* s      = (float*)d_ws;
  float* score2 = s + N;
  float* sortv  = score2 + N;
  int*   sorti  = (int*)(sortv + M);
  int*   nn     = sorti + M;

  // output: x_pool[K*C] | edge_index[2*E] | ew_out[E] | batch_out[K] | perm[K] | topv[K]
  float* out       = (float*)d_out;
  float* out_xpool = out;
  float* out_ei    = out_xpool + (size_t)K * C;
  float* out_ew    = out_ei + 2 * (size_t)E;
  float* out_batch = out_ew + E;
  float* out_perm  = out_batch + K;
  float* out_topv  = out_perm + K;

  // 1) weighted in-degree
  zero_f32_kernel<<<grids(N, THREADS), THREADS, 0, stream>>>(score2, N);
  seg_sum_kernel<<<grids(E, THREADS), THREADS, 0, stream>>>(ew, col, score2, E);

  // 2) WMMA GEMV + blend + tanh -> s[N]   (16 rows per wave32)
  {
    int waves = (N + 15) / 16;
    int wpb = THREADS / 32;
    score_wmma_kernel<<<grids((long long)waves, wpb), THREADS, 0, stream>>>(
        x, p, score2, beta, select_w, s, N, C);
  }

  // 3) exact sorted top-K via global bitonic sort (desc, stable tie-break)
  sort_init_kernel<<<grids(M, THREADS), THREADS, 0, stream>>>(s, sortv, sorti, N, M);
  for (int k = 2; k <= M; k <<= 1)
    for (int j = k >> 1; j > 0; j >>= 1)
      bitonic_step_kernel<<<grids(M, THREADS), THREADS, 0, stream>>>(sortv, sorti, j, k, M);

  // 4) node relabel map
  node_new_init_kernel<<<grids(N, THREADS), THREADS, 0, stream>>>(nn, N);
  node_new_scatter_kernel<<<grids(K, THREADS), THREADS, 0, stream>>>(sorti, nn, K);

  // 5) pooled features, filtered edges, tail outputs
  gather_pool_kernel<<<grids((long long)K * C, THREADS), THREADS, 0, stream>>>(
      x, sortv, sorti, out_xpool, K, C);
  edges_kernel<<<grids(E, THREADS), THREADS, 0, stream>>>(row, col, ew, nn, out_ei, out_ew, E);
  tail_kernel<<<grids(K, THREADS), THREADS, 0, stream>>>(batch, sortv, sorti,
                                                         out_batch, out_perm, out_topv, K);
  (void)n_in; (void)out_size; (void)ws_size;
}